// RateDistortionLoss_S_9740985827725
// MI455X (gfx1250) — compile-verified
//
#include <hip/hip_runtime.h>
#include <math.h>

// ---------------------------------------------------------------------------
// RateDistortionLoss on MI455X (gfx1250, wave32).
// Memory-bound: ~285 MB traffic -> ~12us roofline @ 23.3 TB/s.
//  - float4 streaming loads + global_prefetch_b8 for the big reductions
//  - GLOBAL_LOAD_ASYNC_TO_LDS_B128 (ASYNCcnt) to stage the 48x48 planes for
//    the 5x5 spatial-disentangle box filter (fallback: direct loads)
//  - V_WMMA_F32_16X16X4_F32 vs ones matrix as the wave32 reduction primitive
//  - deterministic two-pass tree reductions (no float atomics)
// ---------------------------------------------------------------------------

typedef __attribute__((ext_vector_type(2))) float v2f;
typedef __attribute__((ext_vector_type(8))) float v8f;
typedef __attribute__((ext_vector_type(4))) int   v4i;

// Sizes from the reference:
#define MSE_N4      7077888   // 16*3*768*768 / 4
#define LIKY_N4     1769472   // 16*192*48*48 / 4
#define LIKZ_N4     110592    // 16*192*12*12 / 4
#define PLANE_HW    2304      // 48*48
#define PW          56        // physical LDS row stride (floats): 16B-aligned interior
#define COLOFF      2         // logical padded col c -> physical col c+COLOFF

#define NB_MSE      1024
#define NB_Y        256
#define NB_Z        54
#define NB_S        3072

#if defined(__has_builtin)
#if __has_builtin(__builtin_amdgcn_global_load_async_to_lds_b128) && \
    __has_builtin(__builtin_amdgcn_s_wait_asynccnt)
#define USE_ASYNC_LDS 1
#endif
#endif

#ifdef USE_ASYNC_LDS
// Builtin signature (from hipcc diagnostic): param0 = v4i in AS1 (__device__),
// param1 = v4i in AS3 (__shared__), then immediate offset + cpol.
typedef __attribute__((address_space(1))) v4i* gv4i_p;
typedef __attribute__((address_space(3))) v4i* lv4i_p;

__device__ __forceinline__ void async_copy_b128(const float* g, float* l) {
  __builtin_amdgcn_global_load_async_to_lds_b128(
      (gv4i_p)(v4i*)const_cast<float*>(g), (lv4i_p)(v4i*)l, 0, 0);
}
#endif

// Wave32 all-lane sum using one f32 WMMA against a ones B-matrix.
// A (16x4) layout: lanes 0-15 supply K=0,1 ; lanes 16-31 supply K=2,3.
// With a = {s, 0}, row m of A holds {s(lane m), 0, s(lane m+16), 0}, so
// D[m][n] = s(m) + s(m+16) for all n. Summing a lane's 8 accumulator VGPRs
// and xor-shuffling across the half-wave boundary yields the full wave sum.
__device__ __forceinline__ float wave_reduce_wmma(float s) {
  v2f a; a[0] = s;   a[1] = 0.0f;
  v2f b; b[0] = 1.0f; b[1] = 1.0f;
  v8f c = {};
  c = __builtin_amdgcn_wmma_f32_16x16x4_f32(
      /*neg_a=*/false, a, /*neg_b=*/false, b,
      /*c_mod=*/(short)0, c, /*reuse_a=*/false, /*reuse_b=*/false);
  float t = c[0] + c[1] + c[2] + c[3] + c[4] + c[5] + c[6] + c[7];
  t += __shfl_xor(t, 16, 32);
  return t;
}

// Block (256 threads = 8 waves) reduction. Result valid in thread 0.
__device__ __forceinline__ float block_reduce_256(float s, float* red) {
  __syncthreads();                 // protect `red` across repeated calls
  float w = wave_reduce_wmma(s);   // EXEC all-ones here: uniform call site
  const int lane = threadIdx.x & 31;
  const int wid  = threadIdx.x >> 5;
  if (lane == 0) red[wid] = w;
  __syncthreads();
  float tot = 0.0f;
  if (threadIdx.x == 0) {
#pragma unroll
    for (int i = 0; i < 8; ++i) tot += red[i];
  }
  return tot;
}

// ----------------------------- MSE partials --------------------------------
__global__ void __launch_bounds__(256)
mse_partial_kernel(const float4* __restrict__ x, const float4* __restrict__ t,
                   float* __restrict__ part, int n4) {
  __shared__ float red[8];
  const int tid    = blockIdx.x * 256 + threadIdx.x;
  const int stride = gridDim.x * 256;
  float s = 0.0f;
  for (int i = tid; i < n4; i += stride) {
    __builtin_prefetch((const void*)(&x[i + stride]), 0, 0);
    __builtin_prefetch((const void*)(&t[i + stride]), 0, 0);
    float4 a = x[i];
    float4 b = t[i];
    float d0 = a.x - b.x, d1 = a.y - b.y, d2 = a.z - b.z, d3 = a.w - b.w;
    s += d0 * d0 + d1 * d1 + d2 * d2 + d3 * d3;
  }
  float tot = block_reduce_256(s, red);
  if (threadIdx.x == 0) part[blockIdx.x] = tot;
}

// --------------------------- log-likelihood partials -----------------------
__global__ void __launch_bounds__(256)
logsum_partial_kernel(const float4* __restrict__ p, float* __restrict__ part, int n4) {
  __shared__ float red[8];
  const int tid    = blockIdx.x * 256 + threadIdx.x;
  const int stride = gridDim.x * 256;
  float s = 0.0f;
  for (int i = tid; i < n4; i += stride) {
    __builtin_prefetch((const void*)(&p[i + stride]), 0, 0);
    float4 a = p[i];
    s += __logf(a.x) + __logf(a.y) + __logf(a.z) + __logf(a.w);
  }
  float tot = block_reduce_256(s, red);
  if (threadIdx.x == 0) part[blockIdx.x] = tot;
}

// ----------------------------- spatial partials ----------------------------
// One 48x48 plane per block. sum_{off != center} |y*shift(y)| ==
// sum_p |y[p]| * (box5x5(|y|)[p] - |y[p]|)   with zero padding.
//
// LDS tile: 52 padded rows x PW(=56) floats. Logical padded col c lives at
// physical col c+COLOFF, so the 48-wide interior starts at physical col 4
// -> every interior row start is 16B-aligned (row stride 224B) for async B128.
__global__ void __launch_bounds__(256)
spatial_partial_kernel(const float* __restrict__ y, float* __restrict__ part) {
  __shared__ float tile[52 * PW];
  __shared__ float red[8];
  const float* base = y + (size_t)blockIdx.x * PLANE_HW;

#ifdef USE_ASYNC_LDS
  // 1) zero only the halo (disjoint from async-written interior -> no race)
  for (int c = threadIdx.x; c < 52 * 52; c += 256) {
    const int r = c / 52, col = c % 52;
    if (r < 2 || r >= 50 || col < 2 || col >= 50)
      tile[r * PW + col + COLOFF] = 0.0f;
  }
  // 2) stage interior rows memory->LDS with async B128 copies (no VGPR data)
  //    48 rows x 12 segments of float4 = 576 transfers
  for (int q = threadIdx.x; q < 48 * 12; q += 256) {
    const int r = q / 12, seg = q % 12;
    async_copy_b128(base + r * 48 + seg * 4, &tile[(r + 2) * PW + 4 + seg * 4]);
  }
  __builtin_amdgcn_s_wait_asynccnt(0);
  __syncthreads();
  // 3) in-place |y| over the interior (exactly 9 cells/thread)
  for (int q = threadIdx.x; q < PLANE_HW; q += 256) {
    const int r = q / 48, col = q % 48;
    const int idx = (r + 2) * PW + 4 + col;
    tile[idx] = fabsf(tile[idx]);
  }
  __syncthreads();
#else
  for (int c = threadIdx.x; c < 52 * 52; c += 256) {
    const int r  = c / 52, col = c % 52;
    const int gr = r - 2,  gc  = col - 2;
    float v = 0.0f;
    if (gr >= 0 && gr < 48 && gc >= 0 && gc < 48) v = fabsf(base[gr * 48 + gc]);
    tile[r * PW + col + COLOFF] = v;
  }
  __syncthreads();
#endif

  float s = 0.0f;
  for (int q = threadIdx.x; q < PLANE_HW; q += 256) {   // exactly 9 iters/thread
    const int r = q / 48, col = q % 48;
    const float a = tile[(r + 2) * PW + (col + 2) + COLOFF];
    float w = 0.0f;
#pragma unroll
    for (int dr = 0; dr < 5; ++dr)
#pragma unroll
      for (int dc = 0; dc < 5; ++dc)
        w += tile[(r + dr) * PW + (col + dc) + COLOFF];
    s += a * (w - a);   // subtract center term
  }
  float tot = block_reduce_256(s, red);
  if (threadIdx.x == 0) part[blockIdx.x] = tot;
}

// ------------------------------- finalize ----------------------------------
__global__ void __launch_bounds__(256)
finalize_kernel(const float* __restrict__ ws, float* __restrict__ out) {
  __shared__ float red[8];
  const float* pm = ws;
  const float* py = pm + NB_MSE;
  const float* pz = py + NB_Y;
  const float* ps = pz + NB_Z;

  float sm = 0.f, sy = 0.f, sz = 0.f, sp = 0.f;
  for (int i = threadIdx.x; i < NB_MSE; i += 256) sm += pm[i];
  for (int i = threadIdx.x; i < NB_Y;   i += 256) sy += py[i];
  for (int i = threadIdx.x; i < NB_Z;   i += 256) sz += pz[i];
  for (int i = threadIdx.x; i < NB_S;   i += 256) sp += ps[i];

  sm = block_reduce_256(sm, red);
  sy = block_reduce_256(sy, red);
  sz = block_reduce_256(sz, red);
  sp = block_reduce_256(sp, red);

  if (threadIdx.x == 0) {
    // inv = 1 / (-ln(2) * N*H*W),  N*H*W = 16*768*768 = 9437184
    const float inv   = (float)(1.0 / (-0.69314718055994530942 * 9437184.0));
    const float y_bpp = sy * inv;
    const float z_bpp = sz * inv;
    const float bpp   = y_bpp + z_bpp;
    const float mse   = sm * (1.0f / 28311552.0f) * 65025.0f;   // mean * 255^2
    const float spat  = sp * (1.0f / 921600.0f);                // / (25*48*48*16)
    const float loss  = 0.01f * mse + bpp + 0.1f * spat;
    out[0] = loss; out[1] = bpp; out[2] = y_bpp;
    out[3] = z_bpp; out[4] = mse; out[5] = spat;
  }
}

// ---------------------------------------------------------------------------
extern "C" void kernel_launch(void* const* d_in, const int* in_sizes, int n_in,
                              void* d_out, int out_size, void* d_ws, size_t ws_size,
                              hipStream_t stream) {
  const float* x_hat  = (const float*)d_in[0];
  const float* target = (const float*)d_in[1];
  const float* y_norm = (const float*)d_in[2];
  const float* lik_y  = (const float*)d_in[3];
  const float* lik_z  = (const float*)d_in[4];
  float* out = (float*)d_out;

  float* ws       = (float*)d_ws;          // needs (1024+256+54+3072)*4 B = ~17.6 KB
  float* part_mse = ws;
  float* part_y   = part_mse + NB_MSE;
  float* part_z   = part_y + NB_Y;
  float* part_s   = part_z + NB_Z;

  mse_partial_kernel<<<NB_MSE, 256, 0, stream>>>(
      (const float4*)x_hat, (const float4*)target, part_mse, MSE_N4);
  logsum_partial_kernel<<<NB_Y, 256, 0, stream>>>(
      (const float4*)lik_y, part_y, LIKY_N4);
  logsum_partial_kernel<<<NB_Z, 256, 0, stream>>>(
      (const float4*)lik_z, part_z, LIKZ_N4);
  spatial_partial_kernel<<<NB_S, 256, 0, stream>>>(y_norm, part_s);
  finalize_kernel<<<1, 256, 0, stream>>>(ws, out);
}